// GCNRegressor_3264175145167
// MI455X (gfx1250) — compile-verified
//
#include <hip/hip_runtime.h>
#include <hip/hip_bf16.h>

typedef __attribute__((ext_vector_type(2))) float v2f;
typedef __attribute__((ext_vector_type(8))) float v8f;

#define HID 128   // EMBED_DIM == HIDDEN == 128

// ---------------------------------------------------------------------------
// Degree / normalization
// ---------------------------------------------------------------------------
__global__ void k_init_deg(float* deg, int n) {
    int i = blockIdx.x * blockDim.x + threadIdx.x;
    if (i < n) deg[i] = 1.0f;               // self-loop contributes 1
}

__global__ void k_accum_deg(const int* __restrict__ dst, float* deg, int nE) {
    int e = blockIdx.x * blockDim.x + threadIdx.x;
    if (e < nE) atomicAdd(&deg[dst[e]], 1.0f);
}

__global__ void k_finalize_dinv(float* deg, int n) {
    int i = blockIdx.x * blockDim.x + threadIdx.x;
    if (i < n) deg[i] = rsqrtf(deg[i]);     // in place: deg -> dinv
}

// ---------------------------------------------------------------------------
// GEMM: out[N x 128] = A[N x 128] @ W[128 x 128] using V_WMMA_F32_16X16X4_F32
//   mode 0: A[row][k] = emb[x[row]][k]          (embedding gather fused)
//   mode 1: A[row][k] = max(Ain[row][k], 0)     (ReLU fused)
// One block = 16 rows; 8 waves each compute one 16x16 output tile (8*16=128 cols).
// K-loop: 128/4 = 32 WMMA steps, fp32 accumulate.
// ---------------------------------------------------------------------------
__global__ __launch_bounds__(256)
void k_gemm_gcn(const float* __restrict__ Ain, const int* __restrict__ xidx,
                const float* __restrict__ emb, const float* __restrict__ W,
                float* __restrict__ out, int n, int mode) {
    __shared__ float lds[16 * HID];         // 8 KB A-tile
    const int row0 = blockIdx.x * 16;
    const int tid  = threadIdx.x;

    // Stage A tile (16 x 128) into LDS; 256 threads x 8 elements.
    for (int i = tid; i < 16 * HID; i += 256) {
        int r = i >> 7, cix = i & (HID - 1);
        int row = row0 + r;
        float v = 0.0f;
        if (row < n) {
            if (mode == 0) {
                v = emb[(size_t)xidx[row] * HID + cix];
            } else {
                v = Ain[(size_t)row * HID + cix];
                v = v > 0.0f ? v : 0.0f;
            }
        }
        lds[i] = v;
    }
    __syncthreads();

    const int wave  = tid >> 5;             // 0..7 -> column tile
    const int lane  = tid & 31;
    const int lrow  = lane & 15;            // M (for A) / N (for B,D)
    const int khalf = (lane >> 4) << 1;     // lanes 16-31 hold K+2,K+3
    const int col0  = wave * 16;

    v8f acc = {};
    for (int k = 0; k < HID; k += 4) {
        v2f a, b;
        // A fragment: 16x4, row = lrow, K = k+khalf, k+khalf+1
        a.x = lds[lrow * HID + k + khalf];
        a.y = lds[lrow * HID + k + khalf + 1];
        // B fragment: 4x16 from W (row-major K x N), col = col0+lrow
        b.x = W[(size_t)(k + khalf) * HID + col0 + lrow];
        b.y = W[(size_t)(k + khalf + 1) * HID + col0 + lrow];
        acc = __builtin_amdgcn_wmma_f32_16x16x4_f32(
            /*neg_a=*/false, a, /*neg_b=*/false, b,
            /*c_mod=*/(short)0, acc, /*reuse_a=*/false, /*reuse_b=*/false);
    }

    // D layout: VGPR r -> M = r (lanes 0-15) / r+8 (lanes 16-31); N = lane&15
    const int mbase = (lane >> 4) << 3;
#pragma unroll
    for (int r = 0; r < 8; ++r) {
        int row = row0 + mbase + r;
        if (row < n) out[(size_t)row * HID + col0 + lrow] = acc[r];
    }
}

// ---------------------------------------------------------------------------
// agg[i][f] = b[f]  (bias pre-seeded so scatter-add lands on top of it)
// ---------------------------------------------------------------------------
__global__ void k_init_bias(float* agg, const float* __restrict__ b, int n) {
    long long t = (long long)blockIdx.x * blockDim.x + threadIdx.x;
    if (t < (long long)n * HID) agg[t] = b[t & (HID - 1)];
}

// ---------------------------------------------------------------------------
// Edge scatter: agg[dst] += hw[src] * dinv[src]*dinv[dst] for E edges + N loops.
// One thread per (edge, 4-feature chunk): float4 gather + 4 float atomics.
// ---------------------------------------------------------------------------
__global__ void k_scatter_edges(const int* __restrict__ src, const int* __restrict__ dst,
                                const float* __restrict__ dinv,
                                const float* __restrict__ hw, float* agg,
                                int nE, int n) {
    long long t = (long long)blockIdx.x * blockDim.x + threadIdx.x;
    int  f4 = (int)(t & 31);
    long long e = t >> 5;
    if (e >= (long long)nE + n) return;
    int s, d;
    if (e < nE) { s = src[e]; d = dst[e]; }
    else        { s = d = (int)(e - nE); }          // self loop
    float w = dinv[s] * dinv[d];
    const float4 hv = *(const float4*)(hw + (size_t)s * HID + f4 * 4);
    float* ap = agg + (size_t)d * HID + f4 * 4;
    atomicAdd(ap + 0, hv.x * w);
    atomicAdd(ap + 1, hv.y * w);
    atomicAdd(ap + 2, hv.z * w);
    atomicAdd(ap + 3, hv.w * w);
}

// ---------------------------------------------------------------------------
// Pooling: sums[g] += relu(agg2[i]); counts[g] += 1
// ---------------------------------------------------------------------------
__global__ void k_zero_pool(float* sums, float* counts, int G) {
    int i = blockIdx.x * blockDim.x + threadIdx.x;
    if (i < G * HID) sums[i] = 0.0f;
    if (i < G) counts[i] = 0.0f;
}

__global__ void k_pool_scatter(const float* __restrict__ agg2, const int* __restrict__ batch,
                               float* sums, float* counts, int n) {
    long long t = (long long)blockIdx.x * blockDim.x + threadIdx.x;
    int f4 = (int)(t & 31);
    long long i = t >> 5;
    if (i >= n) return;
    int g = batch[i];
    float4 hv = *(const float4*)(agg2 + (size_t)i * HID + f4 * 4);
    hv.x = hv.x > 0.0f ? hv.x : 0.0f;
    hv.y = hv.y > 0.0f ? hv.y : 0.0f;
    hv.z = hv.z > 0.0f ? hv.z : 0.0f;
    hv.w = hv.w > 0.0f ? hv.w : 0.0f;
    float* sp = sums + (size_t)g * HID + f4 * 4;
    atomicAdd(sp + 0, hv.x);
    atomicAdd(sp + 1, hv.y);
    atomicAdd(sp + 2, hv.z);
    atomicAdd(sp + 3, hv.w);
    if (f4 == 0) atomicAdd(&counts[g], 1.0f);
}

// ---------------------------------------------------------------------------
// out[g] = (sums[g]/max(count,1)) . fcW + fcb   — one 128-thread block per graph
// ---------------------------------------------------------------------------
__global__ __launch_bounds__(128)
void k_final_fc(const float* __restrict__ sums, const float* __restrict__ counts,
                const float* __restrict__ fcW, const float* __restrict__ fcb,
                float* out, int G) {
    __shared__ float red[HID];
    int g = blockIdx.x;
    int f = threadIdx.x;
    float cnt = counts[g];
    cnt = cnt > 1.0f ? cnt : 1.0f;
    red[f] = (sums[(size_t)g * HID + f] / cnt) * fcW[f];
    __syncthreads();
    for (int s = 64; s > 0; s >>= 1) {
        if (f < s) red[f] += red[f + s];
        __syncthreads();
    }
    if (f == 0) out[g] = red[0] + fcb[0];
}

// ---------------------------------------------------------------------------
extern "C" void kernel_launch(void* const* d_in, const int* in_sizes, int n_in,
                              void* d_out, int out_size, void* d_ws, size_t ws_size,
                              hipStream_t stream) {
    const int*   x     = (const int*)d_in[0];
    const int*   ei    = (const int*)d_in[1];     // [2, E] flat
    const int*   batch = (const int*)d_in[2];
    const float* emb   = (const float*)d_in[3];
    const float* W1    = (const float*)d_in[4];
    const float* b1    = (const float*)d_in[5];
    const float* W2    = (const float*)d_in[6];
    const float* b2    = (const float*)d_in[7];
    const float* fcW   = (const float*)d_in[8];
    const float* fcb   = (const float*)d_in[9];
    float*       out   = (float*)d_out;

    const int N = in_sizes[0];
    const int E = in_sizes[1] / 2;
    const int G = out_size;

    const int* src = ei;
    const int* dst = ei + E;

    // Workspace layout (floats)
    float* buf0   = (float*)d_ws;                 // hw   [N*128]
    float* buf1   = buf0 + (size_t)N * HID;       // agg  [N*128]
    float* dinv   = buf1 + (size_t)N * HID;       // deg -> dinv [N]
    float* sums   = dinv + N;                     // [G*128]
    float* counts = sums + (size_t)G * HID;       // [G]

    const int TPB = 256;
    auto cdiv = [](long long a, long long b) { return (int)((a + b - 1) / b); };

    // 1) normalization coefficients
    k_init_deg     <<<cdiv(N, TPB), TPB, 0, stream>>>(dinv, N);
    k_accum_deg    <<<cdiv(E, TPB), TPB, 0, stream>>>(dst, dinv, E);
    k_finalize_dinv<<<cdiv(N, TPB), TPB, 0, stream>>>(dinv, N);

    const int gemmBlocks    = cdiv(N, 16);
    const long long scatT   = ((long long)E + N) * 32;
    const long long poolT   = (long long)N * 32;

    // 2) layer 1: hw1 = emb[x] @ W1 ; agg1 = b1 + scatter(hw1)
    k_gemm_gcn     <<<gemmBlocks, 256, 0, stream>>>(buf0, x, emb, W1, buf0, N, 0);
    k_init_bias    <<<cdiv((long long)N * HID, TPB), TPB, 0, stream>>>(buf1, b1, N);
    k_scatter_edges<<<cdiv(scatT, TPB), TPB, 0, stream>>>(src, dst, dinv, buf0, buf1, E, N);

    // 3) layer 2: hw2 = relu(agg1) @ W2 ; agg2 = b2 + scatter(hw2)
    k_gemm_gcn     <<<gemmBlocks, 256, 0, stream>>>(buf1, x, emb, W2, buf0, N, 1);
    k_init_bias    <<<cdiv((long long)N * HID, TPB), TPB, 0, stream>>>(buf1, b2, N);
    k_scatter_edges<<<cdiv(scatT, TPB), TPB, 0, stream>>>(src, dst, dinv, buf0, buf1, E, N);

    // 4) mean pool (with relu) + FC
    k_zero_pool    <<<cdiv(G * HID, TPB), TPB, 0, stream>>>(sums, counts, G);
    k_pool_scatter <<<cdiv(poolT, TPB), TPB, 0, stream>>>(buf1, batch, sums, counts, N);
    k_final_fc     <<<G, HID, 0, stream>>>(sums, counts, fcW, fcb, out, G);
}